// MyLSTM_50105088475179
// MI455X (gfx1250) — compile-verified
//
#include <hip/hip_runtime.h>

// LSTM: L=2, H=512, IN=512, B=64, S=1024  (fp32, matches reference precision)
//
// MI455X strategy:
//  - All weights (16.8MB) are L2-resident (192MB L2); HBM traffic is tiny.
//    The recurrence is sync-latency bound -> persistent kernel per layer,
//    16 WGs (1 per WGP, LDS-bound), ONE device barrier per time step.
//  - fp32 WMMA (v_wmma_f32_16x16x4_f32) keeps reference fp32 math.
//  - Software pipelining: the x-half of the K-loop (k<512) is independent of
//    h, so it is computed AFTER signaling barrier arrival (overlapping the
//    wait); only the h-half (k>=512) + gate math sit on the serial chain.
//  - A-staging uses GLOBAL_LOAD_ASYNC_TO_LDS_B128 (ASYNCcnt path), issued
//    before the elementwise phase so the copy overlaps activation math.

#define LNUM 2
#define HD 512
#define IND 512
#define BATCH 64
#define SEQ 1024
#define NG 2048          // 4*HD gate columns
#define KTOT 1024        // IND + HD
#define NWG 16
#define WGCOLS 32        // hidden units per workgroup
#define APAD 1028        // padded LDS row stride (floats) for A
#define NTHREADS 256

typedef __attribute__((ext_vector_type(2))) float v2f;
typedef __attribute__((ext_vector_type(8))) float v8f;
typedef __attribute__((ext_vector_type(4))) int v4i;
typedef __attribute__((address_space(1))) v4i gv4i;   // global int4
typedef __attribute__((address_space(3))) v4i lv4i;   // LDS int4

#if __has_builtin(__builtin_amdgcn_global_load_async_to_lds_b128)
#define HAVE_ASYNC_LDS 1
#else
#define HAVE_ASYNC_LDS 0
#endif

__device__ __forceinline__ void wait_async0() {
#if HAVE_ASYNC_LDS
#if __has_builtin(__builtin_amdgcn_s_wait_asynccnt)
  __builtin_amdgcn_s_wait_asynccnt(0);
#else
  asm volatile("s_wait_asynccnt 0x0" ::: "memory");
#endif
#endif
}

__device__ __forceinline__ void async_cp16(const float* gsrc, float* ldst) {
#if HAVE_ASYNC_LDS
  __builtin_amdgcn_global_load_async_to_lds_b128((gv4i*)gsrc, (lv4i*)ldst, 0, 0);
#else
  *(float4*)ldst = *(const float4*)gsrc;
#endif
}

// One-time prep: zero states + barrier counters, build fused K-major weights
// Wt[l][k][n] (k<512 -> W_i, k>=512 -> W_h) so B-fragment loads are coalesced.
__global__ __launch_bounds__(256)
void lstm_prep_kernel(const float* __restrict__ Wi, const float* __restrict__ Wh,
                      float* __restrict__ Wt, float* __restrict__ h_state,
                      float* __restrict__ c_state, unsigned* __restrict__ ctr) {
  int idx = blockIdx.x * 256 + threadIdx.x;
  if (idx < 8) ctr[idx] = 0u;
  if (idx < LNUM * BATCH * HD) {
    h_state[idx] = 0.0f;
    c_state[idx] = 0.0f;
  }
  if (idx < LNUM * KTOT * NG) {
    int l = idx / (KTOT * NG);
    int rem = idx - l * (KTOT * NG);
    int k = rem / NG;
    int n = rem - k * NG;
    float v = (k < IND) ? Wi[(l * NG + n) * IND + k]
                        : Wh[(l * NG + n) * HD + (k - IND)];
    Wt[idx] = v;  // idx == (l*KTOT + k)*NG + n
  }
}

__device__ __forceinline__ float fast_sigmoid(float x) {
  return 1.0f / (1.0f + __expf(-x));
}
__device__ __forceinline__ float fast_tanh(float x) {
  float e = __expf(2.0f * x);
  return (e - 1.0f) / (e + 1.0f);
}

__global__ __launch_bounds__(NTHREADS, 1)
void lstm_layer_kernel(const float* __restrict__ xin,    // [B,S,512] layer input
                       const float* __restrict__ Wt_l,   // [KTOT,NG]
                       const float* __restrict__ bias_i, // [4,H] this layer
                       const float* __restrict__ bias_h, // [4,H] this layer
                       float* __restrict__ h_state,      // [B,H]
                       float* __restrict__ c_state,      // [B,H]
                       float* __restrict__ hseq_out,     // [B,S,H]
                       float* __restrict__ ht_out,       // [B,H] final h
                       float* __restrict__ ct_out,       // [B,H] final c
                       unsigned* __restrict__ ctr) {
  extern __shared__ float smem[];
  float* ldsA = smem;                    // [64][APAD]  A = [x_s | h_prev]
  float* ldsG = smem + BATCH * APAD;     // [64][128]   gate pre-activations

  const int tid = threadIdx.x;
  const int wg = blockIdx.x;             // 0..15
  const int lane = tid & 31;
  const int wv = tid >> 5;               // 0..7
  const int mrow = (wv >> 1) << 4;       // 0,16,32,48
  const int ngrp = wv & 1;
  const int halfk = lane >> 4;           // K sub-pair select
  const int r = lane & 15;

  // Global column bases of this wave's 4 N-tiles.
  int nbg[4];
#pragma unroll
  for (int tl = 0; tl < 4; ++tl) {
    int t = ngrp * 4 + tl;
    nbg[tl] = (t >> 1) * HD + wg * WGCOLS + (t & 1) * 16;
  }

  // ---- helpers -------------------------------------------------------------
  // Stage x_s into ldsA cols [0,512): 8192 float4 chunks, 32/thread.
  auto stage_x = [&](int s) {
#pragma unroll 4
    for (int i = 0; i < 32; ++i) {
      int chunk = i * NTHREADS + tid;
      int row = chunk >> 7;            // 128 float4 per row-half
      int col = (chunk & 127) << 2;
      async_cp16(xin + (row * SEQ + s) * IND + col, ldsA + row * APAD + col);
    }
  };
  // Stage h_prev into ldsA cols [512,1024).
  auto stage_h = [&]() {
#pragma unroll 4
    for (int i = 0; i < 32; ++i) {
      int chunk = i * NTHREADS + tid;
      int row = chunk >> 7;
      int col = (chunk & 127) << 2;
      async_cp16(h_state + row * HD + col, ldsA + row * APAD + IND + col);
    }
  };
  // Accumulate A[:,kbeg:kbeg+512] x Wt[kbeg:kbeg+512,:] into acc4.
  auto gemm_half = [&](v8f* acc4, int kbeg) {
    const float* pA = ldsA + (mrow + r) * APAD + 2 * halfk;
    const float* pB0 = Wt_l + (2 * halfk) * NG + r;
#pragma unroll 4
    for (int k = kbeg; k < kbeg + HD; k += 4) {
      v2f a;
      a.x = pA[k];
      a.y = pA[k + 1];
      const float* pBk = pB0 + k * NG;
#pragma unroll
      for (int tl = 0; tl < 4; ++tl) {
        v2f b;
        b.x = pBk[nbg[tl]];
        b.y = pBk[nbg[tl] + NG];
        acc4[tl] = __builtin_amdgcn_wmma_f32_16x16x4_f32(
            false, a, false, b, (short)0, acc4[tl], false, false);
      }
    }
  };
  // --------------------------------------------------------------------------

  // Prologue: x-half of step 0 (h-independent).
  v8f acc[4] = {};
  stage_x(0);
  wait_async0();
  __syncthreads();
  gemm_half(acc, 0);

  for (int s = 0; s < SEQ; ++s) {
    // Wait for step s-1's h (all WGs).
    if (s > 0) {
      if (tid == 0) {
        unsigned target = (unsigned)(NWG * s);
        while (__hip_atomic_load(ctr, __ATOMIC_ACQUIRE,
                                 __HIP_MEMORY_SCOPE_AGENT) < target)
          __builtin_amdgcn_s_sleep(2);
      }
      __syncthreads();
      __builtin_amdgcn_fence(__ATOMIC_ACQUIRE, "agent");  // all lanes: inv L0
    }

    // h-half: stage h_prev, accumulate k = 512..1023.
    stage_h();
    wait_async0();
    __syncthreads();
    gemm_half(acc, IND);

    // Spill gate tiles to LDS (C/D layout: vgpr rv -> row mrow + rv + 8*halfk).
#pragma unroll
    for (int tl = 0; tl < 4; ++tl) {
      int t = ngrp * 4 + tl;
      float* gb = ldsG + (mrow + 8 * halfk) * 128 + t * 16 + r;
#pragma unroll
      for (int rv = 0; rv < 8; ++rv) gb[rv * 128] = acc[tl][rv];
    }
    __syncthreads();

    // Prefetch next step's x slab (async copy overlaps elementwise phase;
    // ldsA has no readers until the next gemm_half).
    if (s + 1 < SEQ) stage_x(s + 1);

    // Elementwise LSTM cell update: gate g of local unit hl at ldsG col g*32+hl.
#pragma unroll 2
    for (int i = 0; i < 8; ++i) {
      int e = i * NTHREADS + tid;
      int b = e >> 5;
      int hl = e & 31;
      int hg = wg * WGCOLS + hl;
      const float* gr = ldsG + b * 128 + hl;
      float gi = gr[0]  + bias_i[hg]          + bias_h[hg];
      float gf = gr[32] + bias_i[HD + hg]     + bias_h[HD + hg];
      float gc = gr[64] + bias_i[2 * HD + hg] + bias_h[2 * HD + hg];
      float go = gr[96] + bias_i[3 * HD + hg] + bias_h[3 * HD + hg];
      float it = fast_sigmoid(gi);
      float ft = fast_sigmoid(gf);
      float gt = fast_tanh(gc);
      float ot = fast_sigmoid(go);
      int ci = b * HD + hg;
      float cnew = ft * c_state[ci] + it * gt;
      float hnew = ot * fast_tanh(cnew);
      c_state[ci] = cnew;
      h_state[ci] = hnew;
      hseq_out[(b * SEQ + s) * HD + hg] = hnew;
      if (s == SEQ - 1) {
        ht_out[ci] = hnew;
        ct_out[ci] = cnew;
      }
    }

    // Release h to other WGs, signal arrival (monotonic counter).
    __threadfence();
    __syncthreads();
    if (tid == 0)
      __hip_atomic_fetch_add(ctr, 1u, __ATOMIC_ACQ_REL,
                             __HIP_MEMORY_SCOPE_AGENT);

    // Overlap the barrier wait: x-half of step s+1 (h-independent).
    if (s + 1 < SEQ) {
      v8f accn[4] = {};
      wait_async0();
      __syncthreads();
      gemm_half(accn, 0);
#pragma unroll
      for (int tl = 0; tl < 4; ++tl) acc[tl] = accn[tl];
    }
  }
}

extern "C" void kernel_launch(void* const* d_in, const int* in_sizes, int n_in,
                              void* d_out, int out_size, void* d_ws, size_t ws_size,
                              hipStream_t stream) {
  const float* x   = (const float*)d_in[0];  // [B,S,IN]
  const float* W_i = (const float*)d_in[1];  // [L,4,H,IN]
  const float* W_h = (const float*)d_in[2];  // [L,4,H,H]
  const float* b_i = (const float*)d_in[3];  // [L,4,H]
  const float* b_h = (const float*)d_in[4];  // [L,4,H]

  float* out = (float*)d_out;
  float* hidden_seq = out;                                  // B*S*H
  float* h_t = out + (size_t)BATCH * SEQ * HD;              // L*B*H
  float* c_t = h_t + (size_t)LNUM * BATCH * HD;             // L*B*H

  // Workspace layout
  char* wsb = (char*)d_ws;
  unsigned* ctr  = (unsigned*)wsb;                          // 256 B (2 used)
  float* Wt      = (float*)(wsb + 256);                     // L*KTOT*NG = 16MB
  float* h_state = Wt + (size_t)LNUM * KTOT * NG;           // L*B*H
  float* c_state = h_state + (size_t)LNUM * BATCH * HD;     // L*B*H
  float* h0seq   = c_state + (size_t)LNUM * BATCH * HD;     // B*S*H = 128MB

  {
    long total = (long)LNUM * KTOT * NG;
    int blocks = (int)((total + 255) / 256);
    lstm_prep_kernel<<<blocks, 256, 0, stream>>>(W_i, W_h, Wt, h_state, c_state,
                                                 ctr);
  }

  size_t shmem = (size_t)(BATCH * APAD + BATCH * 128) * sizeof(float); // ~296KB

  // Layer 0: input = x, hidden sequence -> workspace
  lstm_layer_kernel<<<NWG, NTHREADS, shmem, stream>>>(
      x, Wt, b_i, b_h, h_state, c_state, h0seq, h_t, c_t, ctr);

  // Layer 1: input = layer-0 hidden sequence, output -> d_out
  lstm_layer_kernel<<<NWG, NTHREADS, shmem, stream>>>(
      h0seq, Wt + (size_t)KTOT * NG, b_i + NG, b_h + NG,
      h_state + (size_t)BATCH * HD, c_state + (size_t)BATCH * HD, hidden_seq,
      h_t + (size_t)BATCH * HD, c_t + (size_t)BATCH * HD, ctr + 1);
}